// GCN_3702261809372
// MI455X (gfx1250) — compile-verified
//
#include <hip/hip_runtime.h>
#include <hip/hip_bf16.h>

typedef float v2f __attribute__((ext_vector_type(2)));
typedef float v8f __attribute__((ext_vector_type(8)));
typedef unsigned int u32x4 __attribute__((ext_vector_type(4)));
typedef int i32x4 __attribute__((ext_vector_type(4)));
typedef int i32x8 __attribute__((ext_vector_type(8)));

#define Nn 256
#define Cc 64
#define Oo 64
#define Tt 64
#define Vv 25
#define Ss 3
#define Rr 8

#if __has_include(<hip/amd_detail/amd_gfx1250_TDM.h>)
#define TDM_SIX_ARG 1
#endif

// ---------------------------------------------------------------------------
// TDM: async DMA of nElems contiguous f32 from global -> LDS (one row tile).
// D# per CDNA5 ISA ch.8: group0 = {count=1 | lds_addr | global_addr | type=2},
// group1 = {data_size=4B, tensor_dim0=tile_dim0=nElems, tensor_dim1=tile_dim1=1}.
// Issue from ONE wave; producer waits on TENSORcnt, consumers sync at barrier.
// ---------------------------------------------------------------------------
__device__ __forceinline__ void tdm_copy_f32_to_lds(const float* gsrc,
                                                    float* ldst, int nElems) {
#if __has_builtin(__builtin_amdgcn_tensor_load_to_lds)
  const unsigned long long ga = (unsigned long long)(size_t)gsrc;
  const unsigned int la = (unsigned int)(size_t)ldst;  // generic low32 = LDS off
  const unsigned int ne = (unsigned int)nElems;
  u32x4 g0;
  g0[0] = 1u;                                            // count=1, user mode
  g0[1] = la;                                            // lds_addr (bytes)
  g0[2] = (unsigned int)(ga & 0xffffffffu);              // global_addr[31:0]
  g0[3] = (unsigned int)((ga >> 32) & 0x01ffffffu) | (2u << 30);  // [56:32]|type=2
  i32x8 g1;
  g1[0] = (int)(2u << 16);                               // data_size = 4 bytes
  g1[1] = (int)((ne & 0xffffu) << 16);                   // tensor_dim0[15:0]
  g1[2] = (int)((ne >> 16) | (1u << 16));                // tensor_dim0[31:16] | tensor_dim1=1
  g1[3] = (int)((ne & 0xffffu) << 16);                   // tile_dim0
  g1[4] = 1;                                             // tile_dim1 = 1
  g1[5] = (int)ne;                                       // tensor_dim0_stride lo32
  g1[6] = (int)((ne & 0xffffu) << 16);                   // dim1_stride[15:0]
  g1[7] = 0;
  const i32x4 gz = {0, 0, 0, 0};
#ifdef TDM_SIX_ARG
  const i32x8 gz8 = {0, 0, 0, 0, 0, 0, 0, 0};
  __builtin_amdgcn_tensor_load_to_lds(g0, g1, gz, gz, gz8, 0);
#else
  __builtin_amdgcn_tensor_load_to_lds(g0, g1, gz, gz, 0);
#endif
  __builtin_amdgcn_s_wait_tensorcnt(0);
#else
  // Fallback: cooperative copy by this wave only.
  for (int i = (int)(threadIdx.x & 31); i < nElems; i += 32) ldst[i] = gsrc[i];
#endif
}

// ---------------------------------------------------------------------------
// k1: xm[n,c,v] = mean_t x[n,c,t,v].  One block per (n,c); coalesced flat
// reads, LDS atomic accumulation into 25 bins.
// ---------------------------------------------------------------------------
__global__ __launch_bounds__(128, 4) void gcn_k1_xmean(
    const float* __restrict__ x, float* __restrict__ xm) {
  __shared__ float sL[Vv];
  const int nc  = blockIdx.x;
  const int tid = threadIdx.x;
  if (tid < Vv) sL[tid] = 0.0f;
  __syncthreads();
  const float* base = x + (size_t)nc * (Tt * Vv);
  for (int i = tid; i < Tt * Vv; i += 128)
    atomicAdd(&sL[i % Vv], base[i]);
  __syncthreads();
  if (tid < Vv) xm[(size_t)nc * Vv + tid] = sL[tid] * (1.0f / (float)Tt);
}

// ---------------------------------------------------------------------------
// k2: per (n,s): x1[r,u], x2[r,v] (K=64 dots from LDS), then
// y[r,u,v] = tanh(x1[r,u] - x2[r,v]) -> workspace (15.4 MB, stays in L2).
// ---------------------------------------------------------------------------
__global__ __launch_bounds__(256, 2) void gcn_k2_y(
    const float* __restrict__ xm,
    const float* __restrict__ W1, const float* __restrict__ b1,
    const float* __restrict__ W2, const float* __restrict__ b2,
    float* __restrict__ yG) {
  __shared__ float xmL[Cc * Vv];
  __shared__ float W1L[Rr * Cc], W2L[Rr * Cc];
  __shared__ float x1L[Rr * Vv], x2L[Rr * Vv];
  __shared__ float b1L[Rr], b2L[Rr];
  const int s = blockIdx.x, n = blockIdx.y;
  const int tid = threadIdx.x;

  for (int i = tid; i < Cc * Vv; i += 256) xmL[i] = xm[(size_t)n * Cc * Vv + i];
  for (int i = tid; i < Rr * Cc; i += 256) {
    W1L[i] = W1[s * Rr * Cc + i];
    W2L[i] = W2[s * Rr * Cc + i];
  }
  if (tid < Rr) { b1L[tid] = b1[s * Rr + tid]; b2L[tid] = b2[s * Rr + tid]; }
  __syncthreads();

  if (tid < Rr * Vv) {
    const int r = tid / Vv, u = tid % Vv;
    float s1 = b1L[r], s2 = b2L[r];
    #pragma unroll 8
    for (int c = 0; c < Cc; ++c) {
      const float xv = xmL[c * Vv + u];
      s1 = fmaf(W1L[r * Cc + c], xv, s1);
      s2 = fmaf(W2L[r * Cc + c], xv, s2);
    }
    x1L[tid] = s1;
    x2L[tid] = s2;
  }
  __syncthreads();

  const size_t base = (size_t)(n * Ss + s) * (Rr * Vv * Vv);
  for (int i = tid; i < Rr * Vv * Vv; i += 256) {
    const int r = i / (Vv * Vv), rem = i % (Vv * Vv);
    const int u = rem / Vv, v = rem % Vv;
    yG[base + i] = tanhf(x1L[r * Vv + u] - x2L[r * Vv + v]);
  }
}

// ---------------------------------------------------------------------------
// k3: grid = (ob 0..3, n).  Block = 256 threads = 8 waves.  Per s:
//   (a) yL <- y[n,s] via TDM (tensor_load_to_lds + s_wait_tensorcnt)
//   (b) adjB[o'][v][u] = alpha*(W4.y + b4) + PA   (transposed for WMMA-B)
//   (c) x3T = W3[s] @ x[n] + b3 via V_WMMA_F32_16X16X4_F32
//       (M=o' 16, K=c 64, N=1600 cols; B streamed from global w/ one base
//        pointer + immediate offsets; x[n]=410KB so re-reads hit L2)
//   (d) each wave owns (t-tile, u-half), accumulates 16 o-frags
//       D = X3 x adj^T + D in WMMA C regs across s (K=25: 6 clean steps +
//       1 zero-guarded tail step).
// LDS: 37,841 floats = 151.4 KB dynamic.
// ---------------------------------------------------------------------------
__global__ __launch_bounds__(256, 1) void gcn_k3_main(
    const float* __restrict__ x,  const float* __restrict__ PA,
    const float* __restrict__ alpha,
    const float* __restrict__ W3, const float* __restrict__ b3,
    const float* __restrict__ W4, const float* __restrict__ b4,
    const float* __restrict__ yG, float* __restrict__ out) {
  extern __shared__ float sm[];
  float* adjB = sm;                 // 16*25*26 = 10400   [o'][v][u(26)]
  float* x3T  = sm + 10400;         // 16*1600  = 25600   [o'][t*25+v]
  float* yL   = x3T;                // 5000, lifetime-disjoint overlap
  float* W3L  = sm + 36000;         // 16*66    = 1056    [o'][c], stride 66
  float* PAL  = sm + 37056;         // 625                [u][v]
  float* W4L  = sm + 37681;         // 128                [o'][r]
  float* b4L  = sm + 37809;         // 16
  float* b3L  = sm + 37825;         // 16     (total 37841 floats)

  const int ob  = blockIdx.x;
  const int n   = blockIdx.y;
  const int tid = threadIdx.x;
  const int lane = tid & 31, wid = tid >> 5;
  const int lo16 = lane & 15;
  const bool hi  = lane >= 16;
  const int tt = wid & 3;           // this wave's t-tile
  const int uh = wid >> 2;          // this wave's u-half
  const float av = alpha[0];

  const float* xn = x + (size_t)n * (Cc * Tt * Vv);   // [c][t*25+v]

  v8f acc[16];
  const v8f vzero = {};
  #pragma unroll
  for (int o = 0; o < 16; ++o) acc[o] = vzero;

  for (int s = 0; s < Ss; ++s) {
    // -------- (a) per-s loads; y[n,s] via Tensor Data Mover --------
    if (wid == 0) {
      tdm_copy_f32_to_lds(yG + (size_t)(n * Ss + s) * (Rr * Vv * Vv), yL,
                          Rr * Vv * Vv);
    }
    for (int i = tid; i < 16 * Cc; i += 256) {
      const int o = i >> 6, c = i & 63;
      W3L[o * 66 + c] = W3[(s * Oo + ob * 16 + o) * Cc + c];
    }
    for (int i = tid; i < Vv * Vv; i += 256) PAL[i] = PA[s * Vv * Vv + i];
    if (tid < 16 * Rr) W4L[tid] = W4[s * Oo * Rr + ob * 16 * Rr + tid];
    if (tid < 16) {
      b4L[tid] = b4[s * Oo + ob * 16 + tid];
      b3L[tid] = b3[s * Oo + ob * 16 + tid];
    }
    __syncthreads();

    // -------- (b) adjB[o'][v][u] = alpha*(W4.y + b4) + PA --------
    for (int i = tid; i < 16 * Vv * Vv; i += 256) {
      const int o = i / (Vv * Vv), rem = i % (Vv * Vv);
      const int u = rem / Vv, v = rem % Vv;
      float sum = 0.0f;
      #pragma unroll
      for (int r = 0; r < Rr; ++r)
        sum = fmaf(W4L[o * Rr + r], yL[r * Vv * Vv + u * Vv + v], sum);
      adjB[o * (Vv * 26) + v * 26 + u] =
          av * (sum + b4L[o]) + PAL[u * Vv + v];
    }
    __syncthreads();   // adjB done; yL dead -> x3T region reusable

    // -------- (c) stage 1: x3T = W3 @ x + b3 (WMMA) --------
    {
      const float* ap = W3L + lo16 * 66 + (hi ? 2 : 0);     // A via ds imm offs
      for (int ct = wid; ct < 100; ct += 8) {               // 100 col-tiles
        const int j = ct * 16 + lo16;                       // flat col (t*25+v)
        const float* bp = xn + j + (hi ? 2 * 1600 : 0);     // B via global imm offs
        v8f d = vzero;
        #pragma unroll
        for (int k = 0; k < 16; ++k) {                      // K = 64, steps of 4
          v2f a = *(const v2f*)(ap + 4 * k);                // ds_load_b64
          v2f b;
          b.x = bp[k * 6400];                               // imm: k*25600 B
          b.y = bp[k * 6400 + 1600];
          d = __builtin_amdgcn_wmma_f32_16x16x4_f32(
              false, a, false, b, (short)0, d, false, false);
        }
        #pragma unroll
        for (int r = 0; r < 8; ++r) {
          const int o = r + (hi ? 8 : 0);
          x3T[o * 1600 + j] = d[r] + b3L[o];
        }
      }
    }
    __syncthreads();

    // -------- (d) stage 2: OUT += X3 x adj^T (WMMA, accum across s) ------
    {
      const float* x3p = x3T + (tt * 16 + lo16) * Vv + (hi ? 2 : 0);
      const float* adp = adjB + uh * 16 + lo16 + (hi ? 2 * 26 : 0);
      #pragma unroll
      for (int o = 0; o < 16; ++o) {
        #pragma unroll
        for (int k = 0; k < 6; ++k) {                       // clean K-steps
          v2f a, b;
          a.x = x3p[o * 1600 + 4 * k];
          a.y = x3p[o * 1600 + 4 * k + 1];
          b.x = adp[o * 650 + (4 * k) * 26];
          b.y = adp[o * 650 + (4 * k + 1) * 26];
          acc[o] = __builtin_amdgcn_wmma_f32_16x16x4_f32(
              false, a, false, b, (short)0, acc[o], false, false);
        }
        {                                                   // tail K-step: v=24
          v2f a, b;
          a.x = hi ? 0.0f : x3p[o * 1600 + 24];
          a.y = 0.0f;
          b.x = hi ? 0.0f : adp[o * 650 + 24 * 26];
          b.y = 0.0f;
          acc[o] = __builtin_amdgcn_wmma_f32_16x16x4_f32(
              false, a, false, b, (short)0, acc[o], false, false);
        }
      }
    }
    __syncthreads();   // protect adjB/x3T before next s overwrites
  }

  // -------- write out[n, 16*ob+o', t, u] --------
  const int u = uh * 16 + lo16;
  if (u < Vv) {
    #pragma unroll
    for (int o = 0; o < 16; ++o) {
      #pragma unroll
      for (int r = 0; r < 8; ++r) {
        const int t = tt * 16 + r + (hi ? 8 : 0);
        out[(((size_t)(n * Oo + ob * 16 + o)) * Tt + t) * Vv + u] = acc[o][r];
      }
    }
  }
}

// ---------------------------------------------------------------------------
extern "C" void kernel_launch(void* const* d_in, const int* in_sizes, int n_in,
                              void* d_out, int out_size, void* d_ws, size_t ws_size,
                              hipStream_t stream) {
  const float* x     = (const float*)d_in[0];
  const float* PA    = (const float*)d_in[1];
  const float* alpha = (const float*)d_in[2];
  const float* W1    = (const float*)d_in[3];
  const float* b1    = (const float*)d_in[4];
  const float* W2    = (const float*)d_in[5];
  const float* b2    = (const float*)d_in[6];
  const float* W3    = (const float*)d_in[7];
  const float* b3    = (const float*)d_in[8];
  const float* W4    = (const float*)d_in[9];
  const float* b4    = (const float*)d_in[10];
  float* out = (float*)d_out;

  float* xm = (float*)d_ws;                       // 409,600 floats (1.6 MB)
  float* yG = xm + (size_t)Nn * Cc * Vv;          // 3,840,000 floats (15.4 MB)

  gcn_k1_xmean<<<dim3(Nn * Cc), 128, 0, stream>>>(x, xm);
  gcn_k2_y<<<dim3(Ss, Nn), 256, 0, stream>>>(xm, W1, b1, W2, b2, yG);

  const size_t smem = 37841 * sizeof(float);      // 151.4 KB dynamic LDS
  gcn_k3_main<<<dim3(4, Nn), 256, smem, stream>>>(x, PA, alpha, W3, b3, W4, b4,
                                                  yG, out);
}